// Net_19945828122982
// MI455X (gfx1250) — compile-verified
//
#include <hip/hip_runtime.h>
#include <hip/hip_bf16.h>

#define NQ    12
#define DIM   4096         // 2^12
#define BATCH 8

typedef __attribute__((ext_vector_type(2))) float v2f;
typedef __attribute__((ext_vector_type(8))) float v8f;

// ---------------------------------------------------------------------------
// Small-gate helpers operating on a 4096-float state held in LDS.
// Wire w (0 = most significant qubit) maps to bit (NQ-1-w) of the flat index.
// ---------------------------------------------------------------------------
__device__ __forceinline__ void apply_ry_lds(float* s, int bit, float c, float sn,
                                             int tid, int nthreads) {
    const int mask = (1 << bit) - 1;
    for (int p = tid; p < DIM / 2; p += nthreads) {
        int i0 = ((p & ~mask) << 1) | (p & mask);   // bit cleared
        int i1 = i0 | (1 << bit);
        float a0 = s[i0], a1 = s[i1];
        s[i0] = c * a0 - sn * a1;
        s[i1] = sn * a0 + c * a1;
    }
    __syncthreads();
}

__device__ __forceinline__ void apply_cnot_lds(float* s, int cbit, int tbit,
                                               int tid, int nthreads) {
    const int lo = (cbit < tbit) ? cbit : tbit;
    const int hi = (cbit < tbit) ? tbit : cbit;
    const int mlo = (1 << lo) - 1;
    const int mhi = (1 << hi) - 1;
    for (int p = tid; p < DIM / 4; p += nthreads) {
        int x = ((p & ~mlo) << 1) | (p & mlo);      // insert 0 at lo
        x = ((x & ~mhi) << 1) | (x & mhi);          // insert 0 at hi
        int i0 = x | (1 << cbit);                   // ctrl=1, tgt=0
        int i1 = i0 | (1 << tbit);                  // ctrl=1, tgt=1
        float t = s[i0]; s[i0] = s[i1]; s[i1] = t;
    }
    __syncthreads();
}

// ---------------------------------------------------------------------------
// Kernel 1: cir1 (batch-independent) -> v_pre[4096] in workspace
// ---------------------------------------------------------------------------
__global__ __launch_bounds__(256) void cir1_kernel(const float* __restrict__ theta1,
                                                   float* __restrict__ v_out) {
    __shared__ float s[DIM];
    const int tid = threadIdx.x;
    for (int i = tid; i < DIM; i += 256) s[i] = (i == 0) ? 1.0f : 0.0f;
    __syncthreads();

    #pragma unroll
    for (int w = 0; w < NQ; ++w) {
        float t = theta1[w];
        apply_ry_lds(s, NQ - 1 - w, __cosf(t * 0.5f), __sinf(t * 0.5f), tid, 256);
    }
    apply_cnot_lds(s, /*ctrl w1*/ 10, /*tgt w2*/ 9, tid, 256);
    apply_cnot_lds(s, /*ctrl w3*/ 8,  /*tgt w4*/ 7, tid, 256);
    #pragma unroll
    for (int w = 0; w < NQ; ++w) {
        float t = theta1[NQ + w];
        apply_ry_lds(s, NQ - 1 - w, __cosf(t * 0.5f), __sinf(t * 0.5f), tid, 256);
    }
    apply_cnot_lds(s, /*ctrl w0*/ 11, /*tgt w1*/ 10, tid, 256);
    apply_cnot_lds(s, /*ctrl w2*/ 9,  /*tgt w3*/ 8,  tid, 256);

    for (int i = tid; i < DIM; i += 256) v_out[i] = s[i];
}

// ---------------------------------------------------------------------------
// Kernel 2: batched GEMV u[b] = oracles[b] @ v via V_WMMA_F32_16X16X4_F32.
//   * 2 waves per 16-row tile, each covering half of K (2048 cols), to get
//     ~16MB of loads in flight across 4096 waves (latency-BW product for
//     23.3 TB/s @ ~600ns needs ~14MB).
//   * B has v replicated in all 16 columns, so every column of D is the same
//     row-result vector; readout is independent of the N<->lane mapping.
//   * K-slot assignment: lo half-lanes carry cols [kb..kb+3] in slots K0/K1
//     of two chained WMMAs, hi half-lanes carry [kb+4..kb+7] in K2/K3.
//     A and B use the same assignment, so the dot product is exact fp32.
// ---------------------------------------------------------------------------
__global__ __launch_bounds__(256) void oracle_gemv_wmma(
        const float* __restrict__ O,   // (8, 4096, 4096)
        const float* __restrict__ v,   // (4096)
        float* __restrict__ u)         // (8, 4096)
{
    __shared__ float vs[DIM];
    __shared__ float red[8 * 16];      // per-wave 16 partial row sums
    const int b = blockIdx.y;
    const float* A = O + (size_t)b * DIM * DIM;

    // cooperative b128 fill of v into LDS (1024 float4s / 256 threads)
    {
        const float4* v4 = (const float4*)v;
        float4* vs4 = (float4*)vs;
        #pragma unroll
        for (int i = 0; i < 4; ++i) vs4[threadIdx.x + 256 * i] = v4[threadIdx.x + 256 * i];
    }
    __syncthreads();

    const int wave = threadIdx.x >> 5;         // 0..7 (wave32)
    const int lane = threadIdx.x & 31;
    const int tile = wave >> 1;                // 0..3: 16-row tile within block
    const int ksec = wave & 1;                 // 0/1: which K half
    const int m    = lane & 15;                // row within tile
    const int half = lane >> 4;                // 0 = K0/K1 slots, 1 = K2/K3
    const int row0 = (blockIdx.x * 4 + tile) * 16;

    const float* Arow = A + (size_t)(row0 + m) * DIM + ksec * (DIM / 2) + half * 4;
    const float* vrow = vs + ksec * (DIM / 2) + half * 4;

    v8f acc = {};
    for (int kb = 0; kb < DIM / 2; kb += 64) {
        // stream-prefetch ~4KB ahead in this row (speculative; dropped if OOB)
        __builtin_prefetch(Arow + kb + 1024, 0, 0);
        #pragma unroll
        for (int kk = 0; kk < 64; kk += 8) {
            const float4 a  = *(const float4*)(Arow + kb + kk);   // global_load_b128
            const float4 bv = *(const float4*)(vrow + kb + kk);   // ds_load_b128
            v2f a1 = {a.x, a.y}, a2 = {a.z, a.w};
            v2f b1 = {bv.x, bv.y}, b2 = {bv.z, bv.w};
            acc = __builtin_amdgcn_wmma_f32_16x16x4_f32(
                      false, a1, false, b1, (short)0, acc, false, false);
            acc = __builtin_amdgcn_wmma_f32_16x16x4_f32(
                      false, a2, false, b2, (short)0, acc, false, false);
        }
    }

    // D layout: VGPR j -> lanes0-15: M=j (N=lane), lanes16-31: M=8+j.
    // All columns identical; lane 0 / lane 16 hold the 16 partial row sums.
    if (lane == 0) {
        #pragma unroll
        for (int j = 0; j < 8; ++j) red[wave * 16 + j] = acc[j];
    } else if (lane == 16) {
        #pragma unroll
        for (int j = 0; j < 8; ++j) red[wave * 16 + 8 + j] = acc[j];
    }
    __syncthreads();

    // combine the two K-halves; 16 lanes of the even wave store the tile
    if (ksec == 0 && lane < 16) {
        u[(size_t)b * DIM + row0 + lane] =
            red[wave * 16 + lane] + red[(wave + 1) * 16 + lane];
    }
}

// ---------------------------------------------------------------------------
// Kernel 3: cir2 (wires 0..3) + readout -> out[b] = (x, 1-x)
// ---------------------------------------------------------------------------
__global__ __launch_bounds__(256) void cir2_kernel(const float* __restrict__ u,
                                                   const float* __restrict__ theta2,
                                                   float* __restrict__ out) {
    __shared__ float s[DIM];
    const int b = blockIdx.x;
    const int tid = threadIdx.x;
    for (int i = tid; i < DIM; i += 256) s[i] = u[(size_t)b * DIM + i];
    __syncthreads();

    #pragma unroll
    for (int i = 0; i < 4; ++i) {  // wires 0..3 -> bits 11..8
        float t = theta2[i];
        apply_ry_lds(s, 11 - i, __cosf(t * 0.5f), __sinf(t * 0.5f), tid, 256);
    }
    apply_cnot_lds(s, 11, 10, tid, 256);   // CNOT(0,1)
    apply_cnot_lds(s, 9, 8, tid, 256);     // CNOT(2,3)
    #pragma unroll
    for (int i = 0; i < 4; ++i) {
        float t = theta2[4 + i];
        apply_ry_lds(s, 11 - i, __cosf(t * 0.5f), __sinf(t * 0.5f), tid, 256);
    }
    apply_cnot_lds(s, 10, 9, tid, 256);    // CNOT(1,2)

    if (tid == 0) {
        float x = s[0] * s[0] + s[1] * s[1];
        out[b * 2 + 0] = x;
        out[b * 2 + 1] = 1.0f - x;
    }
}

// ---------------------------------------------------------------------------
extern "C" void kernel_launch(void* const* d_in, const int* in_sizes, int n_in,
                              void* d_out, int out_size, void* d_ws, size_t ws_size,
                              hipStream_t stream) {
    const float* oracles = (const float*)d_in[0];   // (8, 4096, 4096) f32
    const float* theta1  = (const float*)d_in[1];   // (24,)
    const float* theta2  = (const float*)d_in[2];   // (8,)
    float* out = (float*)d_out;                     // (8, 2)

    float* ws_v = (float*)d_ws;                     // 4096 floats
    float* ws_u = ws_v + DIM;                       // 8 * 4096 floats

    cir1_kernel<<<1, 256, 0, stream>>>(theta1, ws_v);

    // 256 row-tiles per batch, 4 tiles per block (2 waves each, K split in 2)
    dim3 grid(DIM / (16 * 4), BATCH);               // 64 x 8 blocks, 4096 waves
    oracle_gemv_wmma<<<grid, 256, 0, stream>>>(oracles, ws_v, ws_u);

    cir2_kernel<<<BATCH, 256, 0, stream>>>(ws_u, theta2, out);
}